// PositionalEncoding_15298673509056
// MI455X (gfx1250) — compile-verified
//
#include <hip/hip_runtime.h>
#include <hip/hip_bf16.h>
#include <stdint.h>

// Problem constants (from reference): x is (65536, 64, 16) f32, output = 2*PE
// broadcast over batch. Input is never read.
#define PATCH 16
#define SEQ 64
#define TILE_ELEMS (SEQ * PATCH)            // 1024 f32 = 4096 B per batch tile
#define TILE_BYTES (TILE_ELEMS * 4)
#define THREADS 256
#define TILES_PER_WAVE 8                    // each wave DMA-stores 8 tiles (32 KB)
#define WAVES_PER_BLOCK (THREADS / 32)
#define TILES_PER_BLOCK (TILES_PER_WAVE * WAVES_PER_BLOCK)  // 64 tiles = 256 KB/block

#if defined(__has_builtin)
# if __has_builtin(__builtin_amdgcn_tensor_store_from_lds)
#  define USE_TDM 1
# endif
#endif
#ifndef USE_TDM
# define USE_TDM 0
#endif

typedef __attribute__((ext_vector_type(4))) unsigned int u32x4;
typedef __attribute__((ext_vector_type(8))) int          i32x8;
typedef __attribute__((ext_vector_type(4))) int          i32x4;
typedef __attribute__((ext_vector_type(4))) float        f32x4;

__device__ __forceinline__ float pe_val(int idx) {
  const int i = idx >> 4;   // position 0..63
  const int j = idx & 15;   // channel 0..15
  // div = i / 10000^(j/16*2) = i * exp(-(j/8) * ln(10000))
  const float kLn1e4 = 9.210340371976184f;
  const float div = (float)i * __expf(-(float)j * 0.125f * kLn1e4);
  return 2.0f * (((j & 1) == 0) ? __sinf(div) : __cosf(div));
}

__global__ __launch_bounds__(THREADS)
void PositionalEncoding_15298673509056_kernel(float* __restrict__ out, long long n_tiles) {
  __shared__ float tile[TILE_ELEMS];
  const int tid = threadIdx.x;

  // --- build the 4 KB 2*PE tile once per block (each thread: 4 values) ---
  const int base = tid * 4;
  f32x4 v;
  v.x = pe_val(base + 0);
  v.y = pe_val(base + 1);
  v.z = pe_val(base + 2);
  v.w = pe_val(base + 3);
  *(f32x4*)&tile[base] = v;          // ds_store_b128
  __syncthreads();

#if USE_TDM
  // --- Tensor Data Mover: one iterating descriptor per wave replicates the
  //     LDS tile into TILES_PER_WAVE consecutive batch slots in global memory.
  const int wave = __builtin_amdgcn_readfirstlane(tid >> 5);
  const long long wtile0 =
      (long long)blockIdx.x * TILES_PER_BLOCK + (long long)wave * TILES_PER_WAVE;
  const long long rem = n_tiles - wtile0;
  if (rem > 0) {
    const int cnt = (rem < TILES_PER_WAVE) ? (int)rem : TILES_PER_WAVE;
    const unsigned long long ga =
        (unsigned long long)(uintptr_t)out + (unsigned long long)wtile0 * TILE_BYTES;
    const unsigned int lds_addr =
        (unsigned int)__builtin_amdgcn_readfirstlane((int)(uintptr_t)tile);

    // ---- D# group 0 (128b): count=1 | lds_addr | 57b global_addr | type=2 ----
    u32x4 g0;
    g0[0] = 1u;                                   // count=1, is_restore=0, gather off
    g0[1] = lds_addr;                             // LDS byte address of tile
    g0[2] = (unsigned int)ga;                     // global_addr[31:0]
    g0[3] = ((unsigned int)(ga >> 32) & 0x01FFFFFFu) | 0x80000000u; // addr[56:32] | type=2

    // ---- D# group 1 (256b) ----
    i32x8 g1;
    g1[0] = (int)((2u << 16) | (1u << 19));       // mask=0, data_size=4B, iterate_enable=1
    g1[1] = (int)(16u << 16);                     // atomic_barrier=0 | tensor_dim0[15:0]=16
    g1[2] = 0;                                    // tensor_dim0[31:16]=0 | tensor_dim1[15:0]=0
    g1[3] = (int)(0x0040u | (16u << 16));         // tensor_dim1[31:16]=0x40 (4194304 rows) | tile_dim0=16
    g1[4] = 64;                                   // tile_dim1=64 rows, tile_dim2=0
    g1[5] = 16;                                   // tensor_dim0_stride[31:0]=16 elems (row stride 64 B)
    g1[6] = 0;                                    // stride hi | tensor_dim1_stride lo = 0
    g1[7] = 0;

    // ---- D# group 2 (128b): iteration control ----
    i32x4 g2;
    g2[0] = 1;                                    // tensor_dim2 (unused; tile_dim2=0)
    g2[1] = 0;                                    // lds_addr_increment = 0 -> reuse same tile
    g2[2] = TILE_ELEMS;                           // global_addr_increment = 1024 elems = 4 KB
    g2[3] = (int)(((unsigned int)(cnt - 1)) << 16); // inc hi16=0 | iterate_count = cnt-1

    // ---- D# group 3 (128b): unused dims ----
    i32x4 g3;
    g3[0] = 0; g3[1] = 0; g3[2] = 0; g3[3] = 0;

    // ---- extra operand required by the 6-arg (clang-23 / therock-10.0)
    //      builtin form; architectural D# is fully described by g0..g3 ----
    i32x8 g4;
    g4[0] = 0; g4[1] = 0; g4[2] = 0; g4[3] = 0;
    g4[4] = 0; g4[5] = 0; g4[6] = 0; g4[7] = 0;

    __builtin_amdgcn_tensor_store_from_lds(g0, g1, g2, g3, g4, 0);
    __builtin_amdgcn_s_wait_tensorcnt(0);         // LDS must stay valid until DMA lands
  }
#else
  // --- Fallback: VGPR-resident replication with streaming b128 stores ---
  const long long t0  = (long long)blockIdx.x * TILES_PER_BLOCK;
  long long tend = t0 + TILES_PER_BLOCK;
  if (tend > n_tiles) tend = n_tiles;
  for (long long t = t0; t < tend; ++t) {
    f32x4* dst = (f32x4*)(out + t * TILE_ELEMS) + tid;
    __builtin_nontemporal_store(v, dst);          // global_store_b128, TH=NT
  }
#endif
}

extern "C" void kernel_launch(void* const* d_in, const int* in_sizes, int n_in,
                              void* d_out, int out_size, void* d_ws, size_t ws_size,
                              hipStream_t stream) {
  (void)d_in; (void)in_sizes; (void)n_in; (void)d_ws; (void)ws_size;
  float* out = (float*)d_out;
  const long long n_tiles = (long long)out_size / TILE_ELEMS;   // 65536
  const long long blocks  = (n_tiles + TILES_PER_BLOCK - 1) / TILES_PER_BLOCK; // 1024
  PositionalEncoding_15298673509056_kernel<<<(unsigned)blocks, THREADS, 0, stream>>>(out, n_tiles);
}